// SequenceEncoder_50087908606361
// MI455X (gfx1250) — compile-verified
//
#include <hip/hip_runtime.h>

#define P_N   20000
#define E_N   1280
#define K_N   4096
#define HID_N 1024
#define OUT_N 512

typedef float v2f __attribute__((ext_vector_type(2)));
typedef float v8f __attribute__((ext_vector_type(8)));

// ---- CDNA5 async global->LDS helpers (ASYNCcnt-tracked, bypass VGPRs) ------
__device__ __forceinline__ void async_b128(unsigned lds_off, const float* gsrc) {
    asm volatile("global_load_async_to_lds_b128 %0, %1, off"
                 :: "v"(lds_off), "v"(gsrc) : "memory");
}
__device__ __forceinline__ void async_b32(unsigned lds_off, const float* gsrc) {
    asm volatile("global_load_async_to_lds_b32 %0, %1, off"
                 :: "v"(lds_off), "v"(gsrc) : "memory");
}
template <int N>
__device__ __forceinline__ void wait_asynccnt() {
    asm volatile("s_wait_asynccnt %0" :: "n"(N) : "memory");
}

// ---------------------------------------------------------------------------
// Tiled fp32 WMMA GEMM:  C[M,N] = A[M,K] * op(B) + bias
//   B_IS_NT = true : B stored [N,K] row-major (weight layout), op(B)=B^T
//   B_IS_NT = false: B stored [K,N] row-major,                 op(B)=B
// Block = 256 threads (8 waves), tile 128x64, K-panel 32, double-buffered LDS
// filled by GLOBAL_LOAD_ASYNC_TO_LDS (panel k+1 streams in under panel k's
// WMMAs). Wave w computes rows [w*16,w*16+16) x 64 cols = 4 v8f accumulators.
// ---------------------------------------------------------------------------
#define BM 128
#define BN 64
#define BK 32
#define LPAD 36   // LDS row stride in floats: 144B (16B aligned), conflict-free

template <bool B_IS_NT, bool GUARD_M>
__global__ __launch_bounds__(256) void gemm_wmma_f32(
    const float* __restrict__ A, int lda,
    const float* __restrict__ B, int ldb,
    const float* __restrict__ bias,           // may be nullptr
    float* __restrict__ C, int ldc,
    int M, int N, int K)
{
    __shared__ __align__(16) float As[2 * BM * LPAD];
    __shared__ __align__(16) float Bs[2 * BN * LPAD];

    const int t    = threadIdx.x;
    const int wave = t >> 5;
    const int lane = t & 31;
    const int half = lane >> 4;   // K-half of the fragment
    const int ml   = lane & 15;   // M (or N) index inside fragment

    const int m0 = blockIdx.x * BM;
    const int n0 = blockIdx.y * BN;

    const unsigned asBase = (unsigned)(uintptr_t)&As[0];
    const unsigned bsBase = (unsigned)(uintptr_t)&Bs[0];

    // per-thread async ops per panel (A: 4x b128; B: 2x b128 NT / 8x b32 NN)
    constexpr int NOPS = B_IS_NT ? 6 : 12;

    v8f acc[4] = {};

    auto stage = [&](int k0, int buf) {
        const unsigned aB = asBase + (unsigned)(buf * BM * LPAD * 4);
        const unsigned bB = bsBase + (unsigned)(buf * BN * LPAD * 4);
        #pragma unroll
        for (int i = 0; i < 4; ++i) {          // A panel: BM x BK
            int v   = t + i * 256;
            int row = v >> 3;                  // 8 float4 per row
            int c4  = v & 7;
            int gr  = m0 + row;
            if (GUARD_M && gr >= M) gr = M - 1;
            async_b128(aB + (unsigned)((row * LPAD + c4 * 4) * 4),
                       A + (long)gr * lda + k0 + c4 * 4);
        }
        if (B_IS_NT) {
            #pragma unroll
            for (int i = 0; i < 2; ++i) {      // B panel rows = N, contig in K
                int v   = t + i * 256;
                int row = v >> 3;
                int c4  = v & 7;
                async_b128(bB + (unsigned)((row * LPAD + c4 * 4) * 4),
                           B + (long)(n0 + row) * ldb + k0 + c4 * 4);
            }
        } else {
            #pragma unroll
            for (int i = 0; i < 8; ++i) {      // transpose on stage (coalesced n)
                int e  = t + i * 256;
                int kk = e >> 6;
                int n  = e & 63;
                async_b32(bB + (unsigned)((n * LPAD + kk) * 4),
                          B + (long)(k0 + kk) * ldb + n0 + n);
            }
        }
    };

    const int nPanels = K / BK;
    stage(0, 0);

    for (int it = 0; it < nPanels; ++it) {
        if (it + 1 < nPanels) {
            stage((it + 1) * BK, (it + 1) & 1);  // prefetch next panel
            wait_asynccnt<NOPS>();               // in-order: panel `it` landed
        } else {
            wait_asynccnt<0>();
        }
        __syncthreads();

        const float* Ac = As + (it & 1) * (BM * LPAD);
        const float* Bc = Bs + (it & 1) * (BN * LPAD);
        const int mb = wave * 16;
        #pragma unroll
        for (int kk = 0; kk < BK; kk += 4) {
            const int kf = kk + 2 * half;   // this lane-half holds K = kf, kf+1
            v2f a;
            a.x = Ac[(mb + ml) * LPAD + kf];
            a.y = Ac[(mb + ml) * LPAD + kf + 1];
            #pragma unroll
            for (int j = 0; j < 4; ++j) {
                v2f b;
                b.x = Bc[(j * 16 + ml) * LPAD + kf];
                b.y = Bc[(j * 16 + ml) * LPAD + kf + 1];
                acc[j] = __builtin_amdgcn_wmma_f32_16x16x4_f32(
                    false, a, false, b, (short)0, acc[j], false, false);
            }
        }
        __syncthreads();   // all waves done reading this buffer
    }

    // ---- epilogue: D layout: reg r -> row (wave*16 + r + 8*half), col = j*16+ml
    #pragma unroll
    for (int j = 0; j < 4; ++j) {
        const int col = n0 + j * 16 + ml;
        const float bv = bias ? bias[col] : 0.0f;
        #pragma unroll
        for (int r = 0; r < 8; ++r) {
            int row = m0 + wave * 16 + r + 8 * half;
            if (!GUARD_M || row < M)
                C[(long)row * ldc + col] = acc[j][r] + bv;
        }
    }
}

// ---------------------------------------------------------------------------
// Collapsed biases: b' = W2 @ b1 + b2   (wave-per-output, shuffle reduction)
// waves 0..511 -> protein branch, 512..1023 -> cell branch
// ---------------------------------------------------------------------------
__global__ __launch_bounds__(256) void combine_bias(
    const float* __restrict__ Wp2, const float* __restrict__ bp1, const float* __restrict__ bp2,
    const float* __restrict__ Wc2, const float* __restrict__ bc1, const float* __restrict__ bc2,
    float* __restrict__ bpc, float* __restrict__ bcc)
{
    int g    = blockIdx.x * (blockDim.x >> 5) + (threadIdx.x >> 5);
    int lane = threadIdx.x & 31;
    const float *W, *b1, *b2; float* o; int oi;
    if (g < OUT_N) { W = Wp2; b1 = bp1; b2 = bp2; o = bpc; oi = g; }
    else           { W = Wc2; b1 = bc1; b2 = bc2; o = bcc; oi = g - OUT_N; }
    const float* row = W + (long)oi * HID_N;
    float s = 0.0f;
    for (int h = lane; h < HID_N; h += 32) s += row[h] * b1[h];
    for (int off = 16; off; off >>= 1) s += __shfl_xor(s, off);
    if (lane == 0) o[oi] = s + b2[oi];
}

// ---------------------------------------------------------------------------
// Cell pooling, stage 1: per (cell, K-chunk) partial sums/maxes over column
// pairs. blockDim = 640 (one column-pair per thread), 32 chunks of 128 rows.
// partial layout: [3][32][1280]  (cols 0..639 = sum, 640..1279 = max)
// ---------------------------------------------------------------------------
#define KCHUNK 128
#define NCHUNK (K_N / KCHUNK)   // 32

__global__ __launch_bounds__(640) void pool_partial(
    const float* __restrict__ emb, const int* __restrict__ idx,
    float* __restrict__ partial)
{
    const int cell  = blockIdx.y;
    const int chunk = blockIdx.x;
    const int p     = threadIdx.x;              // pair index 0..639
    const int* ci   = idx + cell * K_N + chunk * KCHUNK;
    float s  = 0.0f;
    float mx = -__builtin_inff();
    for (int k = 0; k < KCHUNK; ++k) {
        const float* row = emb + (long)ci[k] * E_N;
        float2 v = *reinterpret_cast<const float2*>(&row[p * 2]);
        s  += v.x + v.y;
        mx  = fmaxf(mx, fmaxf(v.x, v.y));
    }
    float* out = partial + ((long)(cell * NCHUNK + chunk)) * E_N;
    out[p]       = s;
    out[640 + p] = mx;
}

// stage 2: reduce 32 partials -> concat[3][1280]
__global__ __launch_bounds__(256) void pool_reduce(
    const float* __restrict__ partial, float* __restrict__ concat)
{
    int e = blockIdx.x * blockDim.x + threadIdx.x;
    if (e >= 3 * E_N) return;
    int cell = e / E_N, col = e % E_N;
    const float* p = partial + (long)cell * NCHUNK * E_N + col;
    if (col < 640) {
        float s = 0.0f;
        for (int ch = 0; ch < NCHUNK; ++ch) s += p[ch * E_N];
        concat[e] = s * (1.0f / (float)(K_N * 2));
    } else {
        float m = -__builtin_inff();
        for (int ch = 0; ch < NCHUNK; ++ch) m = fmaxf(m, p[ch * E_N]);
        concat[e] = m;
    }
}

// Final cell GEMV: out[3][512] = concat[3][1280] @ Wc'[512,1280]^T + bc'
__global__ __launch_bounds__(256) void cell_gemv(
    const float* __restrict__ concat, const float* __restrict__ W,
    const float* __restrict__ bias, float* __restrict__ out)
{
    int g    = blockIdx.x * (blockDim.x >> 5) + (threadIdx.x >> 5); // 0..1535
    int lane = threadIdx.x & 31;
    int cell = g / OUT_N, o = g % OUT_N;
    const float* x = concat + (long)cell * E_N;
    const float* w = W + (long)o * E_N;
    float s = 0.0f;
    for (int e = lane; e < E_N; e += 32) s += x[e] * w[e];
    for (int off = 16; off; off >>= 1) s += __shfl_xor(s, off);
    if (lane == 0) out[cell * OUT_N + o] = s + bias[o];
}

// ---------------------------------------------------------------------------
extern "C" void kernel_launch(void* const* d_in, const int* in_sizes, int n_in,
                              void* d_out, int out_size, void* d_ws, size_t ws_size,
                              hipStream_t stream)
{
    const float* emb = (const float*)d_in[0];   // [20000,1280]
    const int*   idx = (const int*)  d_in[1];   // [3,4096]
    const float* Wp1 = (const float*)d_in[2];   // [1024,1280]
    const float* bp1 = (const float*)d_in[3];
    const float* Wp2 = (const float*)d_in[4];   // [512,1024]
    const float* bp2 = (const float*)d_in[5];
    const float* Wc1 = (const float*)d_in[6];
    const float* bc1 = (const float*)d_in[7];
    const float* Wc2 = (const float*)d_in[8];
    const float* bc2 = (const float*)d_in[9];

    float* out_prot = (float*)d_out;                     // [20000,512]
    float* out_cell = (float*)d_out + (long)P_N * OUT_N; // [3,1,512]

    // workspace layout (floats)
    float* ws   = (float*)d_ws;
    float* wcp  = ws;                              // [512,1280] collapsed protein W
    float* wcc  = wcp + (long)OUT_N * E_N;         // [512,1280] collapsed cell W
    float* bpc  = wcc + (long)OUT_N * E_N;         // [512]
    float* bcc  = bpc + OUT_N;                     // [512]
    float* part = bcc + OUT_N;                     // [3,32,1280]
    float* cat  = part + 3L * NCHUNK * E_N;        // [3,1280]

    // 1) collapsed biases (independent of everything else)
    combine_bias<<<128, 256, 0, stream>>>(Wp2, bp1, bp2, Wc2, bc1, bc2, bpc, bcc);

    // 2) collapse weights: Wcomb[512,1280] = W2[512,1024] @ W1[1024,1280]  (NN)
    dim3 gw(OUT_N / BM, E_N / BN);   // (4,20)
    gemm_wmma_f32<false, false><<<gw, 256, 0, stream>>>(
        Wp2, HID_N, Wp1, E_N, nullptr, wcp, E_N, OUT_N, E_N, HID_N);
    gemm_wmma_f32<false, false><<<gw, 256, 0, stream>>>(
        Wc2, HID_N, Wc1, E_N, nullptr, wcc, E_N, OUT_N, E_N, HID_N);

    // 3) main GEMM: out[20000,512] = X[20000,1280] @ Wcomb^T + b'   (NT)
    dim3 gm((P_N + BM - 1) / BM, OUT_N / BN);   // (157,8)
    gemm_wmma_f32<true, true><<<gm, 256, 0, stream>>>(
        emb, E_N, wcp, E_N, bpc, out_prot, OUT_N, P_N, OUT_N, E_N);

    // 4) cell branch: gather+pool partials, reduce, tiny GEMV
    pool_partial<<<dim3(NCHUNK, 3), 640, 0, stream>>>(emb, idx, part);
    pool_reduce<<<(3 * E_N + 255) / 256, 256, 0, stream>>>(part, cat);
    cell_gemv<<<192, 256, 0, stream>>>(cat, wcc, bcc, out_cell);
}